// LSTM_38328288150270
// MI455X (gfx1250) — compile-verified
//
#include <hip/hip_runtime.h>
#include <cmath>

typedef __attribute__((ext_vector_type(2))) float v2f;
typedef __attribute__((ext_vector_type(8))) float v8f;

#define T_STEPS 65536
#define HID     512
#define VOC     256
#define G4H     2048   // 4*HID
#define CHUNK   4096
#define NWG     16
#define WSTRIDE 516    // padded LDS row stride for W_hh slice (conflict avoidance)

__device__ __forceinline__ v8f wmma_f32(v2f a, v2f b, v8f c) {
  // V_WMMA_F32_16X16X4_F32: 8-arg form (neg_a, A, neg_b, B, c_mod, C, reuse_a, reuse_b)
  return __builtin_amdgcn_wmma_f32_16x16x4_f32(false, a, false, b, (short)0, c, false, false);
}

// C[M x N] = A[M x K] * B[N x K]^T + bias[N], all row-major fp32.
// Block: 256 threads = 8 waves as 2(M) x 4(N); block tile 32(M) x 64(N); K staged in 64-chunks.
__global__ __launch_bounds__(256) void gemm_bias_wmma(
    const float* __restrict__ A, const float* __restrict__ B,
    const float* __restrict__ bias, float* __restrict__ C,
    int K, int N)
{
  __shared__ float As[32 * 66];   // [m][k], padded
  __shared__ float Bs[64 * 65];   // transposed: [k][n], padded

  const int tid  = threadIdx.x;
  const int lane = tid & 31;
  const int wave = tid >> 5;
  const int wm   = wave >> 2;     // 0..1 (M)
  const int wn   = wave & 3;      // 0..3 (N)
  const int l15  = lane & 15;
  const int lhi  = lane >> 4;     // 0 or 1
  const int m0   = blockIdx.y * 32;
  const int n0   = blockIdx.x * 64;

  v8f acc = {};

  for (int k0 = 0; k0 < K; k0 += 64) {
    // stage A tile 32x64 (coalesced over k)
    for (int e = tid; e < 32 * 64; e += 256) {
      const int r = e >> 6, k = e & 63;
      As[r * 66 + k] = A[(size_t)(m0 + r) * K + (k0 + k)];
    }
    // stage B tile transposed: Bs[k][n] = B[n0+n][k0+k]
    for (int e = tid; e < 64 * 64; e += 256) {
      const int n = e >> 6, k = e & 63;
      Bs[k * 65 + n] = B[(size_t)(n0 + n) * K + (k0 + k)];
    }
    __syncthreads();
    #pragma unroll
    for (int kk = 0; kk < 64; kk += 4) {
      // A frag 16x4: lanes 0-15 hold K={kk,kk+1}, lanes 16-31 hold K={kk+2,kk+3}
      v2f a = *(const v2f*)&As[(wm * 16 + l15) * 66 + kk + 2 * lhi];
      // B frag 4x16: VGPR0 = K (kk|kk+1), VGPR1 = K (kk+2|kk+3), N striped over lanes
      v2f b;
      b.x = Bs[(kk + lhi) * 65 + wn * 16 + l15];
      b.y = Bs[(kk + 2 + lhi) * 65 + wn * 16 + l15];
      acc = wmma_f32(a, b, acc);
    }
    __syncthreads();
  }

  // epilogue: C/D layout — VGPR r -> M = r + 8*(lane>=16), N = lane&15
  const int n  = n0 + wn * 16 + l15;
  const float bv = bias[n];
  const int mb = m0 + wm * 16 + lhi * 8;
  #pragma unroll
  for (int r = 0; r < 8; ++r)
    C[(size_t)(mb + r) * N + n] = acc[r] + bv;
}

// Persistent 16-workgroup LSTM scan. Workgroup wg owns h elements [32*wg, 32*wg+32)
// and its 128 W_hh rows (4 gates x 32) staged in LDS (~258 KB). c lives in LDS.
// h_t exchanged through Hall (L2-resident row) guarded by a global atomic barrier.
__global__ __launch_bounds__(256) void lstm_scan(
    const float* __restrict__ Zc,     // [steps][2048] = W_ih@x + b for this chunk
    const float* __restrict__ W_hh,   // [2048][512]
    const float* __restrict__ h0, const float* __restrict__ c0,
    float* __restrict__ Hall,         // [T][512]
    float* __restrict__ c_buf,        // [512] carry between chunk launches
    unsigned int* __restrict__ cnt,   // global barrier counter (monotonic)
    float* __restrict__ out_hc,       // d_out + T*V : final h (512) then c (512)
    int t0, int steps, int first_chunk)
{
  extern __shared__ float smem[];
  float* W_s  = smem;                     // 128 * WSTRIDE
  float* h_s  = smem + 128 * WSTRIDE;     // 512
  float* pa_s = h_s + 512;                // 128 partial sums
  float* zb_s = pa_s + 128;               // 128 gate pre-activations
  float* c_s  = zb_s + 128;               // 32 cell states

  const int tid  = threadIdx.x;
  const int wg   = blockIdx.x;            // 0..15
  const int half = tid >> 7;              // K half: 0 -> [0,256), 1 -> [256,512)
  const int r    = tid & 127;             // local row 0..127
  const int gate = r >> 5;                // 0:i 1:f 2:g 3:o
  const int jloc = r & 31;
  const int grow = gate * HID + wg * 32 + jloc;   // row in [0,2048)

  // stage this WGP's W_hh slice into LDS (once per chunk)
  for (int e = tid; e < 128 * HID; e += 256) {
    const int rr = e >> 9, k = e & 511;
    const int gr = (rr >> 5) * HID + wg * 32 + (rr & 31);
    W_s[rr * WSTRIDE + k] = W_hh[(size_t)gr * HID + k];
  }
  if (tid < 32) c_s[tid] = first_chunk ? c0[wg * 32 + tid] : c_buf[wg * 32 + tid];
  __syncthreads();

  const int koff = half * 256;

  for (int lt = 0; lt < steps; ++lt) {
    const int t = t0 + lt;

    // broadcast h_{t-1} into LDS (barrier at end of previous step guarantees visibility)
    const float* hsrc = (t == 0) ? h0 : (Hall + (size_t)(t - 1) * HID);
    h_s[tid]       = hsrc[tid];
    h_s[tid + 256] = hsrc[tid + 256];
    __syncthreads();

    float zval = 0.0f;
    if (half == 0) {
      zval = Zc[(size_t)lt * G4H + grow];
      if (lt + 1 < steps)
        __builtin_prefetch(&Zc[(size_t)(lt + 1) * G4H + grow], 0, 0);
    }

    // half-row matvec: 256 MACs, float4 from LDS, 4 accumulators for ILP
    const float* wr = &W_s[r * WSTRIDE + koff];
    const float* hp = &h_s[koff];
    float s0 = 0.f, s1 = 0.f, s2 = 0.f, s3 = 0.f;
    #pragma unroll 16
    for (int k = 0; k < 256; k += 4) {
      const float4 w4 = *(const float4*)(wr + k);
      const float4 h4 = *(const float4*)(hp + k);
      s0 = fmaf(w4.x, h4.x, s0);
      s1 = fmaf(w4.y, h4.y, s1);
      s2 = fmaf(w4.z, h4.z, s2);
      s3 = fmaf(w4.w, h4.w, s3);
    }
    const float sum = (s0 + s1) + (s2 + s3);
    if (half == 1) pa_s[r] = sum;
    __syncthreads();
    if (half == 0) zb_s[r] = sum + pa_s[r] + zval;
    __syncthreads();

    if (tid < 32) {
      const float zi = zb_s[tid], zf = zb_s[32 + tid];
      const float zg = zb_s[64 + tid], zo = zb_s[96 + tid];
      const float ig = 1.0f / (1.0f + expf(-zi));
      const float fg = 1.0f / (1.0f + expf(-zf));
      const float gg = tanhf(zg);
      const float og = 1.0f / (1.0f + expf(-zo));
      const float cn = fg * c_s[tid] + ig * gg;
      c_s[tid] = cn;
      const float hn = og * tanhf(cn);
      const int j = wg * 32 + tid;
      Hall[(size_t)t * HID + j] = hn;
      if (t == T_STEPS - 1) { out_hc[j] = hn; out_hc[HID + j] = cn; }
      __threadfence();   // make h_t agent-visible before barrier arrive
    }
    __syncthreads();

    if (tid == 0) {
      __hip_atomic_fetch_add(cnt, 1u, __ATOMIC_RELEASE, __HIP_MEMORY_SCOPE_AGENT);
      const unsigned int target = 16u * (unsigned int)(t + 1);
      while (__hip_atomic_load(cnt, __ATOMIC_ACQUIRE, __HIP_MEMORY_SCOPE_AGENT) < target)
        __builtin_amdgcn_s_sleep(1);
    }
    __syncthreads();
  }

  if (tid < 32) c_buf[wg * 32 + tid] = c_s[tid];
}

extern "C" void kernel_launch(void* const* d_in, const int* in_sizes, int n_in,
                              void* d_out, int out_size, void* d_ws, size_t ws_size,
                              hipStream_t stream)
{
  const float* X      = (const float*)d_in[0];   // [T][256]
  const float* h0     = (const float*)d_in[1];   // [512]
  const float* c0     = (const float*)d_in[2];   // [512]
  const float* W_ih   = (const float*)d_in[3];   // [2048][256]
  const float* W_hh   = (const float*)d_in[4];   // [2048][512]
  const float* b      = (const float*)d_in[5];   // [2048]
  const float* W_head = (const float*)d_in[6];   // [256][512]
  const float* b_head = (const float*)d_in[7];   // [256]
  float* out = (float*)d_out;                    // [T][256] ++ h_f[512] ++ c_f[512]

  char* ws = (char*)d_ws;
  unsigned int* cnt = (unsigned int*)ws;                         // 256 B
  float* c_buf = (float*)(ws + 256);                             // 2 KB
  float* Zbuf  = (float*)(ws + 4096);                            // CHUNK*2048 f32 = 32 MB
  float* Hall  = (float*)(ws + 4096 + (size_t)CHUNK * G4H * 4);  // T*512 f32 = 128 MB

  hipMemsetAsync(cnt, 0, 256, stream);   // reset barrier counter (graph node, deterministic)

  const size_t smem_scan = (size_t)(128 * WSTRIDE + 512 + 128 + 128 + 32) * sizeof(float);

  for (int c = 0; c < T_STEPS / CHUNK; ++c) {
    // Phase 1 (chunked): Z = X_chunk @ W_ih^T + b   (fp32 WMMA)
    dim3 gA(G4H / 64, CHUNK / 32);
    gemm_bias_wmma<<<gA, 256, 0, stream>>>(X + (size_t)c * CHUNK * VOC, W_ih, b,
                                           Zbuf, VOC, G4H);
    // Phase 2: sequential scan over this chunk (persistent, 16 WGPs, W_hh in LDS)
    lstm_scan<<<NWG, 256, smem_scan, stream>>>(Zbuf, W_hh, h0, c0, Hall, c_buf, cnt,
                                               out + (size_t)T_STEPS * VOC,
                                               c * CHUNK, CHUNK, c == 0);
  }

  // Phase 3: Y = Hall @ W_head^T + b_head   (fp32 WMMA)
  dim3 gH(VOC / 64, T_STEPS / 32);
  gemm_bias_wmma<<<gH, 256, 0, stream>>>(Hall, W_head, b_head, out, HID, VOC);
}